// MappingNetwork_41781441856238
// MI455X (gfx1250) — compile-verified
//
#include <hip/hip_runtime.h>

// ---------------------------------------------------------------------------
// MappingNetwork for MI455X (gfx1250).
// B=8192, LAT=64, STYLE=64, NDOM=8, H=512.
//
//  * Bucket samples by domain -> unshared branches computed once per sample
//    (8x less work than reference's compute-all-then-gather).
//  * fp32 WMMA V_WMMA_F32_16X16X4_F32 (exact fp32 math).
//  * 256-thread WGs (8 wave32s), WG tile 128x64, wave tile 64x16.
//  * NEW: tiles are fed with GLOBAL_LOAD_ASYNC_TO_LDS_B128 (ASYNCcnt path,
//    no VGPR staging); A keeps the stride-36 swizzle via per-lane LDS dests,
//    W is stored k-major with row stride 80 dwords (conflict-free B frags).
// ---------------------------------------------------------------------------

#define BATCH 8192
#define LAT   64
#define STYLE 64
#define NDOM  8
#define HID   512

typedef __attribute__((ext_vector_type(2))) float v2f;
typedef __attribute__((ext_vector_type(8))) float v8f;

// ---------------- small bucketing kernels ----------------------------------
__global__ void k_zero_cnt(int* cnt) {
    if (threadIdx.x < NDOM) cnt[threadIdx.x] = 0;
}

__global__ void k_count(const int* __restrict__ y, int* __restrict__ cnt,
                        int* __restrict__ ranks) {
    int b = blockIdx.x * blockDim.x + threadIdx.x;
    ranks[b] = atomicAdd(&cnt[y[b]], 1);
}

__global__ void k_scan(const int* __restrict__ cnt, int* __restrict__ off) {
    int s = 0;
    for (int d = 0; d < NDOM; ++d) { off[d] = s; s += cnt[d]; }
}

__global__ void k_scatter(const int* __restrict__ y, const int* __restrict__ ranks,
                          const int* __restrict__ off, int* __restrict__ perm) {
    int b = blockIdx.x * blockDim.x + threadIdx.x;
    perm[off[y[b]] + ranks[b]] = b;
}

// ---------------- async global->LDS helpers --------------------------------
__device__ __forceinline__ void async_b128(unsigned lds_off, const float* gaddr) {
    asm volatile("global_load_async_to_lds_b128 %0, %1, off"
                 :: "v"(lds_off), "v"(gaddr) : "memory");
}
__device__ __forceinline__ void async_wait0() {
    asm volatile("s_wait_asynccnt 0" ::: "memory");
}

// ---------------- fused GEMM + bias + ReLU ---------------------------------
// SEG:     blockIdx.x encodes (domain, m-block); rows come from a domain
//          segment [off[d], off[d]+cnt[d]) of the grouped buffer.
// GATHER:  A row g is fetched from A[perm[g]] (entry into unshared branches).
// SCATTER: output row g is written to Out[perm[g]] (final layer -> d_out).
#define KC      32
#define ASTR    36                       // LDS A row stride (floats)
#define WSTR    80                       // LDS W row stride (floats)
#define A_SZ    (128 * ASTR)             // 4608 floats
#define W_SZ    (KC * WSTR)              // 2560 floats
#define BUF_SZ  (A_SZ + W_SZ)            // 7168 floats

template <int SEG, int GATHER, int SCATTER>
__global__ __launch_bounds__(256)
void gemm_bias_relu(const float* __restrict__ A, const float* __restrict__ W,
                    const float* __restrict__ bias, float* __restrict__ Out,
                    int K, int N, int ldA,
                    const int* __restrict__ perm, const int* __restrict__ cnt,
                    const int* __restrict__ off, int domBlocks) {
    __shared__ float lds[2 * BUF_SZ];

    const int t    = threadIdx.x;
    const int lane = t & 31;
    const int w    = t >> 5;            // wave id 0..7
    const int bx   = blockIdx.x;
    const int n0blk = blockIdx.y * 64;

    int count = 0, segOff = 0, mb128 = 0, blockRow = 0;
    if (SEG) {
        const int dom = bx / domBlocks;
        const int mb  = bx % domBlocks;
        count = cnt[dom];
        mb128 = mb * 128;
        if (mb128 >= count) return;      // uniform exit: EXEC stays all-1s
        segOff = off[dom];
        W    += (size_t)dom * K * N;
        bias += (size_t)dom * N;
    } else {
        blockRow = bx * 128;
    }

    // ---- per-thread global source pointers (fixed across K-chunks) ----
    const float* aPtr[4];
#pragma unroll
    for (int i = 0; i < 4; ++i) {
        const int ml = (t >> 3) + i * 32;            // local row 0..127
        int arow;
        if (SEG) {
            int lr = mb128 + ml;
            if (lr > count - 1) lr = count - 1;       // clamp (masked on store)
            const int g = segOff + lr;
            arow = GATHER ? perm[g] : g;
        } else {
            arow = blockRow + ml;
        }
        aPtr[i] = A + (size_t)arow * ldA + ((t & 7) << 2);
    }
    const float* wPtr = W + n0blk + ((t & 15) << 2);
    const int    wRow = t >> 4;                      // 0..15

    // ---- per-thread LDS destination offsets (bytes) ----
    const unsigned ldsBase = (unsigned)(unsigned long long)(&lds[0]);
    unsigned aDst[4], wDst[2];
#pragma unroll
    for (int i = 0; i < 4; ++i)
        aDst[i] = ldsBase + (((t >> 3) + i * 32) * ASTR + ((t & 7) << 2)) * 4u;
#pragma unroll
    for (int i = 0; i < 2; ++i)
        wDst[i] = ldsBase + (A_SZ + (wRow + i * 16) * WSTR + ((t & 15) << 2)) * 4u;

    auto issueChunk = [&](int k0, int buf) {
        const unsigned bofs = (unsigned)buf * (BUF_SZ * 4u);
#pragma unroll
        for (int i = 0; i < 4; ++i)
            async_b128(aDst[i] + bofs, aPtr[i] + k0);
#pragma unroll
        for (int i = 0; i < 2; ++i)
            async_b128(wDst[i] + bofs, wPtr + (size_t)(k0 + wRow + i * 16) * N);
    };

    // ---- accumulators: wave owns 4 M-tiles x 1 N-tile ----
    v8f c[4] = {};
    const int r     = lane & 15;
    const int half  = lane >> 4;
    const int mBase = (w >> 2) * 64;                 // 0 or 64
    const int nRow  = (w & 3) * 16 + r;              // 0..63
    const int kOff  = half * 2;                      // lanes16-31 take k+2,k+3

    auto compute = [&](int buf) {
        const float* lA = lds + buf * BUF_SZ;
        const float* lW = lA + A_SZ;
#pragma unroll
        for (int kk = 0; kk < KC; kk += 4) {
            const float* p = lW + (kk + kOff) * WSTR + nRow;
            v2f bf = { p[0], p[WSTR] };
#pragma unroll
            for (int i = 0; i < 4; ++i) {
                v2f af = *(const v2f*)&lA[(mBase + i * 16 + r) * ASTR + kk + kOff];
                c[i] = __builtin_amdgcn_wmma_f32_16x16x4_f32(
                    false, af, false, bf, (short)0, c[i], false, false);
            }
        }
    };

    // ---- double-buffered main loop (async copies overlap WMMA) ----
    const int nChunks = K / KC;
    issueChunk(0, 0);
    async_wait0();
    __syncthreads();
    for (int g = 0; g < nChunks; ++g) {
        if (g + 1 < nChunks) issueChunk((g + 1) * KC, (g + 1) & 1);
        compute(g & 1);
        async_wait0();                   // own async writes for g+1 done
        __syncthreads();                 // published to all waves
    }

    // ---- epilogue: bias + ReLU + (scatter) store ----
    const int coln = n0blk + (w & 3) * 16 + r;
    const float bv = bias[coln];
#pragma unroll
    for (int i = 0; i < 4; ++i) {
#pragma unroll
        for (int v = 0; v < 8; ++v) {
            const int ml = mBase + i * 16 + half * 8 + v;   // 0..127
            int orow; bool ok = true;
            if (SEG) {
                const int lr = mb128 + ml;
                ok = lr < count;
                const int g = segOff + lr;
                orow = SCATTER ? (ok ? perm[g] : 0) : g;
            } else {
                orow = blockRow + ml;
            }
            if (ok)
                Out[(size_t)orow * N + coln] = fmaxf(c[i][v] + bv, 0.0f);
        }
    }
}

// ---------------------------------------------------------------------------
extern "C" void kernel_launch(void* const* d_in, const int* in_sizes, int n_in,
                              void* d_out, int out_size, void* d_ws, size_t ws_size,
                              hipStream_t stream) {
    const float* z   = (const float*)d_in[0];
    const int*   y   = (const int*)  d_in[1];
    const float* sw0 = (const float*)d_in[2];  const float* sb0 = (const float*)d_in[3];
    const float* sw1 = (const float*)d_in[4];  const float* sb1 = (const float*)d_in[5];
    const float* sw2 = (const float*)d_in[6];  const float* sb2 = (const float*)d_in[7];
    const float* sw3 = (const float*)d_in[8];  const float* sb3 = (const float*)d_in[9];
    const float* uw0 = (const float*)d_in[10]; const float* ub0 = (const float*)d_in[11];
    const float* uw1 = (const float*)d_in[12]; const float* ub1 = (const float*)d_in[13];
    const float* uw2 = (const float*)d_in[14]; const float* ub2 = (const float*)d_in[15];
    const float* uw3 = (const float*)d_in[16]; const float* ub3 = (const float*)d_in[17];
    float* out = (float*)d_out;

    // workspace layout (~33.6 MB)
    float* xa    = (float*)d_ws;
    float* xb    = xa + (size_t)BATCH * HID;
    int*   perm  = (int*)(xb + (size_t)BATCH * HID);
    int*   ranks = perm + BATCH;
    int*   cnt   = ranks + BATCH;
    int*   off   = cnt + NDOM;

    // ---- bucket samples by domain ----
    k_zero_cnt<<<1, 32, 0, stream>>>(cnt);
    k_count  <<<BATCH / 256, 256, 0, stream>>>(y, cnt, ranks);
    k_scan   <<<1, 1, 0, stream>>>(cnt, off);
    k_scatter<<<BATCH / 256, 256, 0, stream>>>(y, ranks, off, perm);

    const dim3 blk(256);
    const int  MB   = BATCH / 128;   // 64 m-blocks
    const dim3 gS(MB, HID / 64);     // shared layers: 64 x 8
    const dim3 gU(NDOM * MB, HID / 64);
    const dim3 gF(NDOM * MB, STYLE / 64);

    // ---- shared MLP (ping-pong xa/xb) ----
    gemm_bias_relu<0,0,0><<<gS, blk, 0, stream>>>(z,  sw0, sb0, xa, LAT, HID, LAT,
                                                  nullptr, nullptr, nullptr, 0);
    gemm_bias_relu<0,0,0><<<gS, blk, 0, stream>>>(xa, sw1, sb1, xb, HID, HID, HID,
                                                  nullptr, nullptr, nullptr, 0);
    gemm_bias_relu<0,0,0><<<gS, blk, 0, stream>>>(xb, sw2, sb2, xa, HID, HID, HID,
                                                  nullptr, nullptr, nullptr, 0);
    gemm_bias_relu<0,0,0><<<gS, blk, 0, stream>>>(xa, sw3, sb3, xb, HID, HID, HID,
                                                  nullptr, nullptr, nullptr, 0);

    // ---- unshared branches on domain-grouped rows ----
    gemm_bias_relu<1,1,0><<<gU, blk, 0, stream>>>(xb, uw0, ub0, xa, HID, HID, HID,
                                                  perm, cnt, off, MB);     // gather
    gemm_bias_relu<1,0,0><<<gU, blk, 0, stream>>>(xa, uw1, ub1, xb, HID, HID, HID,
                                                  perm, cnt, off, MB);
    gemm_bias_relu<1,0,0><<<gU, blk, 0, stream>>>(xb, uw2, ub2, xa, HID, HID, HID,
                                                  perm, cnt, off, MB);
    gemm_bias_relu<1,0,1><<<gF, blk, 0, stream>>>(xa, uw3, ub3, out, HID, STYLE, HID,
                                                  perm, cnt, off, MB);     // scatter
    (void)in_sizes; (void)n_in; (void)out_size; (void)ws_size;
}